// StateProbabilitiesPaulied_22024592293958
// MI455X (gfx1250) — compile-verified
//
#include <hip/hip_runtime.h>
#include <hip/hip_bf16.h>
#include <math.h>

#define BATCH        131072
#define NDRIVES      16
#define NPAULI       64
#define DIM          8
#define HN           128      // 2*DIM*DIM floats per H (re/im interleaved)
#define SLAB_S       132      // padded slab row stride (floats) to break bank conflicts
#define RS           17       // padded 16x16 tile row stride
#define WAVES_PER_WG 4
#define NTERMS       12
#define MAX_SQ       24

typedef __attribute__((ext_vector_type(2))) float v2f;
typedef __attribute__((ext_vector_type(8))) float v8f;

// Compiler barrier + hardware DS-counter drain: makes cross-lane LDS
// communication within a wave safe without workgroup barriers.
__device__ __forceinline__ void lds_fence() {
  asm volatile("s_wait_dscnt 0" ::: "memory");
}

// 16x16x16 real matmul, A operand preloaded in registers (4 chunks of v2f),
// B operand streamed from an LDS tile with row stride RS.
// A chunk kk: lane l<16 holds A[l][4kk],A[l][4kk+1]; lanes 16-31 hold
// A[l-16][4kk+2..3]. B chunk kk: lane n<16 holds B[4kk][n],B[4kk+1][n];
// lanes 16-31 hold B[4kk+2][n],B[4kk+3][n]. D: (m,n) at lane n+16*(m/8), vgpr m%8.
__device__ __forceinline__ v8f wmma_mm16_regA(const v2f* a4, const float* Bm, int lane) {
  const int rc   = lane & 15;
  const int off2 = (lane >> 4) << 1;
  v8f acc = {};
#pragma unroll
  for (int kk = 0; kk < 4; ++kk) {
    const int ka = 4 * kk + off2;
    v2f b;
    b.x = Bm[ka * RS + rc];
    b.y = Bm[(ka + 1) * RS + rc];
    acc = __builtin_amdgcn_wmma_f32_16x16x4_f32(false, a4[kk], false, b, (short)0, acc,
                                                false, false);
  }
  return acc;
}

// Same matmul but with the A operand also streamed from LDS (used when A
// changes every iteration, i.e. the squaring loop).
__device__ __forceinline__ v8f wmma_mm16(const float* A, const float* Bm, int lane) {
  const int rc   = lane & 15;
  const int off2 = (lane >> 4) << 1;
  v2f a4[4];
#pragma unroll
  for (int kk = 0; kk < 4; ++kk) {
    const int ka = 4 * kk + off2;
    a4[kk].x = A[rc * RS + ka];
    a4[kk].y = A[rc * RS + ka + 1];
  }
  return wmma_mm16_regA(a4, Bm, lane);
}

// Fold kernel & bias through the Pauli basis once:
//   Wg[d][m] = sum_k kernel[d][k] * Pflat[k][m]   (16 x 128)
//   Cg[m]    = sum_k bias[k]      * Pflat[k][m]   (128)
// where m = (i*8+j)*2 + (0:re,1:im) indexes the flattened complex 8x8 basis.
__global__ void prep_kernel(const float* __restrict__ kern,
                            const float* __restrict__ bias,
                            const double* __restrict__ paul,
                            float* __restrict__ Wg, float* __restrict__ Cg) {
  const int m = threadIdx.x;           // 0..127
  const int e = m >> 1, c = m & 1;
  float acc[NDRIVES];
#pragma unroll
  for (int d = 0; d < NDRIVES; ++d) acc[d] = 0.f;
  float cb = 0.f;
  for (int k = 0; k < NPAULI; ++k) {
    const float p = (float)paul[(size_t)(k * 64 + e) * 2 + c];
    cb += bias[k] * p;
#pragma unroll
    for (int d = 0; d < NDRIVES; ++d) acc[d] += kern[d * NPAULI + k] * p;
  }
#pragma unroll
  for (int d = 0; d < NDRIVES; ++d) Wg[d * HN + m] = acc[d];
  Cg[m] = cb;
}

__global__ __launch_bounds__(128)
void state_probs_kernel(const float* __restrict__ x,
                        const float* __restrict__ Wg,
                        const float* __restrict__ Cg,
                        float* __restrict__ out) {
  __shared__ float Wlds[NDRIVES * HN];              // 8 KB, shared by all waves
  __shared__ float Clds[HN];
  __shared__ float slab[WAVES_PER_WG][16 * SLAB_S]; // per-wave 16x128 H slab
  __shared__ float Xb[WAVES_PER_WG][16 * RS];       // scaled real-form A
  __shared__ float Rb[WAVES_PER_WG][16 * RS];       // running result / term

  const int tid  = threadIdx.x;
  const int wid  = tid >> 5;
  const int lane = tid & 31;
  const int col  = lane & 15;
  const int half = lane >> 4;
  const int off2 = half << 1;

  // Stage weights once per workgroup (only uniform barrier in the kernel).
  for (int i = tid; i < NDRIVES * HN; i += 128) Wlds[i] = Wg[i];
  if (tid < HN) Clds[tid] = Cg[tid];
  __syncthreads();

  const int rowBase = (blockIdx.x * WAVES_PER_WG + wid) * 16;

  // ---- Phase 1: H slab = x_tile(16x16) @ W(16x128) + c, via WMMA ----
  v2f a4[4];
#pragma unroll
  for (int kk = 0; kk < 4; ++kk) {
    const float* xr = x + (size_t)(rowBase + col) * NDRIVES + 4 * kk + off2;
    a4[kk].x = xr[0];
    a4[kk].y = xr[1];
  }
#pragma unroll
  for (int t = 0; t < 8; ++t) {
    v8f acc;
    const float bv = Clds[t * 16 + col];   // bias depends only on column
#pragma unroll
    for (int r = 0; r < 8; ++r) acc[r] = bv;
#pragma unroll
    for (int kk = 0; kk < 4; ++kk) {
      const int kr = 4 * kk + off2;
      v2f b;
      b.x = Wlds[kr * HN + t * 16 + col];
      b.y = Wlds[(kr + 1) * HN + t * 16 + col];
      acc = __builtin_amdgcn_wmma_f32_16x16x4_f32(false, a4[kk], false, b, (short)0,
                                                  acc, false, false);
    }
#pragma unroll
    for (int r = 0; r < 8; ++r)
      slab[wid][(r + 8 * half) * SLAB_S + t * 16 + col] = acc[r];
  }
  lds_fence();

  // ---- Phase 2: per matrix, exp(-iH)|0> by scaling & squaring (all WMMA) ----
  float* X = Xb[wid];
  float* R = Rb[wid];
  for (int mb = 0; mb < 16; ++mb) {
    const float* hrow = &slab[wid][mb * SLAB_S];
    // Real-form of A = -iH: [[Hi, Hr], [-Hr, Hi]]  (16x16), track max |entry|.
    float pmax = 0.f;
#pragma unroll
    for (int t = 0; t < 8; ++t) {
      const int e = lane + 32 * t;
      const int r = e >> 4, c = e & 15;
      const int i = r & 7, j = c & 7;
      const float hre = hrow[(i * 8 + j) * 2 + 0];
      const float him = hrow[(i * 8 + j) * 2 + 1];
      const float v = (r < 8) ? ((c < 8) ? him : hre)
                              : ((c < 8) ? -hre : him);
      X[r * RS + c] = v;
      pmax = fmaxf(pmax, fabsf(v));
    }
    // Wave-wide max via LDS (R reused as 32-float scratch); uniform result
    // -> uniform squaring count s for the whole wave.
    R[lane] = pmax;
    lds_fence();
    float nm = 0.f;
#pragma unroll
    for (int l = 0; l < 32; ++l) nm = fmaxf(nm, R[l]);
    const float nrm = nm * 16.0f;            // row-sum norm bound
    int s = 0;
    if (nrm > 1.0f) {
      s = (int)ceilf(log2f(nrm));
      if (s > MAX_SQ) s = MAX_SQ;
    }
    const float sc = exp2f(-(float)s);
#pragma unroll
    for (int t = 0; t < 8; ++t) {
      const int e = lane + 32 * t;
      X[(e >> 4) * RS + (e & 15)] *= sc;
    }
    // R = I
#pragma unroll
    for (int t = 0; t < 8; ++t) {
      const int e = lane + 32 * t;
      R[(e >> 4) * RS + (e & 15)] = ((e >> 4) == (e & 15)) ? 1.f : 0.f;
    }
    lds_fence();

    // X is invariant across the whole Taylor recursion: load its A-operand
    // register image once (survives the asm memory clobbers; only the B
    // operand must be re-streamed from LDS each iteration).
    v2f xa[4];
#pragma unroll
    for (int kk = 0; kk < 4; ++kk) {
      const int ka = 4 * kk + off2;
      xa[kk].x = X[col * RS + ka];
      xa[kk].y = X[col * RS + ka + 1];
    }

    // Horner Taylor: R <- I + (X*R)/n, n = NTERMS..1  => R ~ exp(X/2^s)
    for (int n = NTERMS; n >= 1; --n) {
      const v8f acc = wmma_mm16_regA(xa, R, lane);
      const float inv = 1.0f / (float)n;
      lds_fence();
#pragma unroll
      for (int r = 0; r < 8; ++r) {
        const int m = r + 8 * half;
        R[m * RS + col] = acc[r] * inv + ((m == col) ? 1.f : 0.f);
      }
      lds_fence();
    }
    // Squarings: R <- R*R, s times (s uniform across the wave).
    for (int it = 0; it < s; ++it) {
      const v8f acc = wmma_mm16(R, R, lane);
      lds_fence();
#pragma unroll
      for (int r = 0; r < 8; ++r)
        R[(r + 8 * half) * RS + col] = acc[r];
      lds_fence();
    }
    // amp[j] = E[j][0]: Er = R[j][0], Ei = R[j+8][0]; out = |amp|^2.
    if (lane < DIM) {
      const float er = R[lane * RS + 0];
      const float ei = R[(lane + 8) * RS + 0];
      out[(size_t)(rowBase + mb) * DIM + lane] = er * er + ei * ei;
    }
    lds_fence();
  }
}

extern "C" void kernel_launch(void* const* d_in, const int* in_sizes, int n_in,
                              void* d_out, int out_size, void* d_ws, size_t ws_size,
                              hipStream_t stream) {
  const float*  x    = (const float*)d_in[0];
  const float*  kern = (const float*)d_in[1];
  const float*  bias = (const float*)d_in[2];
  const double* paul = (const double*)d_in[3];  // complex128 interleaved

  float* Wg  = (float*)d_ws;          // 16*128 floats
  float* Cg  = Wg + NDRIVES * HN;     // 128 floats
  float* out = (float*)d_out;

  prep_kernel<<<1, 128, 0, stream>>>(kern, bias, paul, Wg, Cg);
  state_probs_kernel<<<BATCH / (16 * WAVES_PER_WG), 128, 0, stream>>>(x, Wg, Cg, out);
}